// SDPANet_62156766708218
// MI455X (gfx1250) — compile-verified
//
#include <hip/hip_runtime.h>

// ---------------------------------------------------------------------------
// SDPANet (involution-style dynamic conv) for gfx1250 / MI455X, wave32.
//
//   wdyn = conv2( relu( BN( conv1(x) ) ) )          -- WMMA f32 16x16x4 GEMMs
//   out[b,c,p] = sum_k unfold3x3(x)[b,c,k,p] * wdyn[b,k,p]   -- VALU stencil
//
// Output reshape (b, h*w, c) is a raw reinterpret of (b,c,h,w)-contiguous,
// so kernel 2 writes out in (b,c,h,w) row-major order directly.
// ---------------------------------------------------------------------------

typedef __attribute__((ext_vector_type(2))) float v2f;
typedef __attribute__((ext_vector_type(8))) float v8f;

#define B_DIM  8
#define C_IN   256
#define C_R    64
#define H_DIM  64
#define W_DIM  64
#define HW     4096      // 64*64
#define KK     9         // 3x3
#define BN_EPS 1e-5f

#if __has_builtin(__builtin_amdgcn_wmma_f32_16x16x4_f32)
#define HAVE_WMMA_F32X4 1
#else
#define HAVE_WMMA_F32X4 0
#endif

__device__ __forceinline__ v8f wmma_f32x4(v2f a, v2f b, v8f c) {
#if HAVE_WMMA_F32X4
    // 8 args: (neg_a, A, neg_b, B, c_mod, C, reuse_a, reuse_b)
    return __builtin_amdgcn_wmma_f32_16x16x4_f32(false, a, false, b, (short)0, c,
                                                 false, false);
#else
    // Placeholder so the TU still compiles if the builtin is missing; the
    // histogram will show wmma==0 and we pivot to the bf16 16x16x32 path.
    c[0] += a.x * b.x;
    return c;
#endif
}

// ---------------------------------------------------------------------------
// Kernel 1: per 16-pixel tile, Y[64x16] = W1[64x256] @ X[256x16] (+BN+ReLU),
// then Wdyn[9x16] = W2[9x64] @ Y[64x16] (+bias).  One wave32 per block.
//
// WMMA f32 16x16x4 fragment layout (ISA 7.12.2):
//   A (16x4):  lanes 0-15 -> M=lane, vgpr0=K0,vgpr1=K1 ; lanes 16-31 -> K2,K3
//   B (4x16):  lanes 0-15 -> N=lane, vgpr0=K0,vgpr1=K1 ; lanes 16-31 -> K2,K3
//   C/D (16x16): vgpr r, lanes 0-15 -> M=r, N=lane ; lanes 16-31 -> M=r+8
// ---------------------------------------------------------------------------
__global__ __launch_bounds__(32) void sdpanet_wdyn_kernel(
    const float* __restrict__ x,     // (B, C_IN, H, W)
    const float* __restrict__ w1,    // (C_R, C_IN)
    const float* __restrict__ b1,    // (C_R)
    const float* __restrict__ gma,   // (C_R)
    const float* __restrict__ bta,   // (C_R)
    const float* __restrict__ mu,    // (C_R)
    const float* __restrict__ var,   // (C_R)
    const float* __restrict__ w2,    // (KK, C_R)
    const float* __restrict__ b2,    // (KK)
    float* __restrict__ wdyn)        // (B, KK, HW)
{
    __shared__ float ylds[C_R * 16];              // Y tile, 4 KB

    const int lane  = threadIdx.x;                // 0..31, wave32
    const int col   = lane & 15;                  // N (pixel) / M (row) index
    const int half  = lane >> 4;                  // 0: K0/K1, 1: K2/K3

    const int tile    = blockIdx.x;               // 8 * 256 tiles
    const int b       = tile >> 8;
    const int pixbase = (tile & 255) << 4;
    const int pix     = pixbase + col;

    const float* xb = x + (size_t)b * C_IN * HW;

    // ---------------- GEMM1: Y = W1 @ X ----------------
    v8f acc[4];
    #pragma unroll
    for (int mt = 0; mt < 4; ++mt) acc[mt] = v8f{};

    #pragma unroll 4
    for (int k = 0; k < C_IN; k += 4) {
        const int kk = k + (half << 1);           // this lane's K pair base
        v2f bf;                                   // B frag: X[kk..kk+1][pix]
        bf.x = xb[(size_t)kk * HW + pix];
        bf.y = xb[(size_t)(kk + 1) * HW + pix];
        #pragma unroll
        for (int mt = 0; mt < 4; ++mt) {
            const int row = (mt << 4) + col;      // output channel
            v2f af;                               // A frag: W1[row][kk..kk+1]
            af.x = w1[row * C_IN + kk];
            af.y = w1[row * C_IN + kk + 1];
            acc[mt] = wmma_f32x4(af, bf, acc[mt]);
        }
    }

    // -------- bias + BN(eval) + ReLU, stage Y to LDS --------
    #pragma unroll
    for (int mt = 0; mt < 4; ++mt) {
        #pragma unroll
        for (int r = 0; r < 8; ++r) {
            const int ch = (mt << 4) + r + (half << 3);   // channel of this D row
            const float scale = gma[ch] * rsqrtf(var[ch] + BN_EPS);
            float v = acc[mt][r] + b1[ch];
            v = (v - mu[ch]) * scale + bta[ch];
            v = fmaxf(v, 0.0f);
            ylds[ch * 16 + col] = v;
        }
    }
    __syncthreads();

    // ---------------- GEMM2: Wdyn = W2 @ Y ----------------
    v8f accw = v8f{};
    const int orow = col;                          // W2 row (pad 9 -> 16 with 0)
    #pragma unroll
    for (int k = 0; k < C_R; k += 4) {
        const int kk = k + (half << 1);
        v2f af;
        af.x = (orow < KK) ? w2[orow * C_R + kk]     : 0.0f;
        af.y = (orow < KK) ? w2[orow * C_R + kk + 1] : 0.0f;
        v2f bf;
        bf.x = ylds[kk * 16 + col];
        bf.y = ylds[(kk + 1) * 16 + col];
        accw = wmma_f32x4(af, bf, accw);
    }

    // -------- store wdyn (+bias); guard divergence AFTER all WMMAs --------
    #pragma unroll
    for (int r = 0; r < 8; ++r) {
        const int o = r + (half << 3);
        if (o < KK) {
            wdyn[((size_t)b * KK + o) * HW + pix] = accw[r] + b2[o];
        }
    }
}

// ---------------------------------------------------------------------------
// Kernel 2: out[b,c,h,w] = sum_{i,j} x[b,c,h+i-1,w+j-1] * wdyn[b,3i+j,h,w]
// Memory bound: coalesced along w for x loads, wdyn loads, out stores.
// ---------------------------------------------------------------------------
__global__ __launch_bounds__(256) void sdpanet_apply_kernel(
    const float* __restrict__ x,     // (B, C_IN, H, W)
    const float* __restrict__ wdyn,  // (B, KK, HW)
    float* __restrict__ out)         // (B, C_IN, HW) == (b, h*w, c) reinterpret
{
    const int gid = blockIdx.x * 256 + threadIdx.x;   // over B*C_IN*HW
    const int pix = gid & (HW - 1);
    const int bc  = gid >> 12;
    const int c   = bc & (C_IN - 1);
    const int b   = bc >> 8;
    const int h   = pix >> 6;
    const int w   = pix & (W_DIM - 1);

    const float* wd = wdyn + (size_t)b * KK * HW + pix;
    const float* xc = x + ((size_t)b * C_IN + c) * HW;

    float s = 0.0f;
    #pragma unroll
    for (int i = 0; i < 3; ++i) {
        const int hh = h + i - 1;
        #pragma unroll
        for (int j = 0; j < 3; ++j) {
            const int ww = w + j - 1;
            const bool inb = (hh >= 0) & (hh < H_DIM) & (ww >= 0) & (ww < W_DIM);
            const float xv = inb ? xc[hh * W_DIM + ww] : 0.0f;
            s = fmaf(xv, wd[(i * 3 + j) * HW], s);
        }
    }
    out[gid] = s;
}

// ---------------------------------------------------------------------------
extern "C" void kernel_launch(void* const* d_in, const int* in_sizes, int n_in,
                              void* d_out, int out_size, void* d_ws, size_t ws_size,
                              hipStream_t stream) {
    const float* x      = (const float*)d_in[0];
    const float* w1     = (const float*)d_in[1];
    const float* b1     = (const float*)d_in[2];
    const float* gma    = (const float*)d_in[3];
    const float* bta    = (const float*)d_in[4];
    const float* mu     = (const float*)d_in[5];
    const float* var    = (const float*)d_in[6];
    const float* w2     = (const float*)d_in[7];
    const float* b2     = (const float*)d_in[8];
    float* out          = (float*)d_out;
    float* wdyn         = (float*)d_ws;           // B*KK*HW floats = 1.18 MB

    // Kernel 1: 8 batches * 256 pixel-tiles, one wave32 each.
    sdpanet_wdyn_kernel<<<B_DIM * (HW / 16), 32, 0, stream>>>(
        x, w1, b1, gma, bta, mu, var, w2, b2, wdyn);

    // Kernel 2: one thread per output element.
    const int total = B_DIM * C_IN * HW;
    sdpanet_apply_kernel<<<total / 256, 256, 0, stream>>>(x, wdyn, out);
}